// ContrastiveCenterLoss_26620207301028
// MI455X (gfx1250) — compile-verified
//
#include <hip/hip_runtime.h>

// ContrastiveCenterLoss, factored form for MI455X (gfx1250, wave32).
//
// loss = (1/2B) * intra / (inter + 1e-6) / 0.1
//   Sf       = sum_b ||f_b||^2
//   Sc       = sum_c ||c_c||^2
//   crossAll = sum_b (f_b . sumC)          where sumC = sum_c c_c
//   SqL      = sum_b ||c_{label[b]}||^2
//   G        = sum_b  f_b . c_{label[b]}   <-- V_WMMA_F32_16X16X4_F32 diag trick
//   total = C*Sf + B*Sc - 2*crossAll ; intra = Sf + SqL - 2*G ; inter = total - intra

typedef __attribute__((ext_vector_type(2))) float v2f;
typedef __attribute__((ext_vector_type(8))) float v8f;

#define B_ROWS 16384
#define C_ROWS 4096
#define D_DIM  128

// workspace layout (float offsets)
#define WS_COLPART 0       // [32][128] center column-sum partials
#define WS_SQPART  4096    // [32]      center sq-sum partials
#define WS_SUMC    4160    // [128]     final column sums of centers
#define WS_SC      4290    // [1]       Sc
#define WS_FEATP   4320    // [128][4]  feat-pass partials: sf, sql, cross, g

// ---- Pass 1: per-WG partial column sums + squared sums of centers ----------
__global__ void ccl_centers_partial(const float* __restrict__ centers,
                                    float* __restrict__ ws) {
  const int w  = blockIdx.x;      // 32 WGs, 128 center rows each
  const int t  = threadIdx.x;     // 256 threads
  const int d  = t & 127;
  const int rr = t >> 7;          // 0/1: two rows in flight per column
  float colsum = 0.f, sq = 0.f;
  const int base_row = w * 128;
  #pragma unroll 4
  for (int i = 0; i < 64; ++i) {
    float v = centers[(size_t)(base_row + rr + 2 * i) * D_DIM + d];
    colsum += v;
    sq     += v * v;
  }
  __shared__ float lds_col[256];
  lds_col[t] = colsum;
  __syncthreads();
  if (t < 128) ws[WS_COLPART + w * 128 + t] = lds_col[t] + lds_col[t + 128];

  // reduce sq over the workgroup (wave32 shuffle tree, then LDS across 8 waves)
  for (int off = 16; off; off >>= 1) sq += __shfl_xor(sq, off, 32);
  __shared__ float lds_sq[8];
  const int lane = t & 31, wv = t >> 5;
  if (lane == 0) lds_sq[wv] = sq;
  __syncthreads();
  if (t == 0) {
    float s = 0.f;
    for (int i = 0; i < 8; ++i) s += lds_sq[i];
    ws[WS_SQPART + w] = s;
  }
}

// ---- Pass 2: fixed-order reduce of center partials -> sumC[128], Sc --------
__global__ void ccl_centers_reduce(float* __restrict__ ws) {
  const int t = threadIdx.x;      // 128 threads
  if (t < 128) {
    float s = 0.f;
    for (int w = 0; w < 32; ++w) s += ws[WS_COLPART + w * 128 + t];
    ws[WS_SUMC + t] = s;
  }
  if (t == 0) {
    float s = 0.f;
    for (int w = 0; w < 32; ++w) s += ws[WS_SQPART + w];
    ws[WS_SC] = s;
  }
}

// ---- Pass 3: feat streaming pass; WMMA for the gathered batched dot --------
// 128 WGs x 256 threads = 1024 waves; each wave owns one 16-row tile of feat.
// Per K-chunk of 4: A(16x4)=feat tile, B(4x16)=gathered centers.
// A/B VGPR layout (half-wave K-offset pattern): lane L loads the contiguous
// float2 at column (kb + (L>=16)*2) of row (L&15) -> one global_load_b64 each.
// D accumulates over all K; diag(D)[m] = f_m . c_{label[m]} extracted once.
__global__ void ccl_feat_pass(const float* __restrict__ feat,
                              const int*   __restrict__ label,
                              const float* __restrict__ centers,
                              float* __restrict__ ws) {
  __shared__ float s_sumC[D_DIM];
  if (threadIdx.x < D_DIM) s_sumC[threadIdx.x] = ws[WS_SUMC + threadIdx.x];
  __syncthreads();

  const int lane  = threadIdx.x & 31;
  const int wv    = threadIdx.x >> 5;
  const int m     = lane & 15;
  const int khalf = lane >> 4;                    // 0 or 1
  const int gwid  = blockIdx.x * 8 + wv;          // 0..1023
  const int nblocks = B_ROWS / 16;                // 1024
  const int nwaves  = gridDim.x * 8;

  v8f acc = {};                                   // persists across whole K loop
  float sf = 0.f, sql = 0.f, cross = 0.f;

  for (int blk = gwid; blk < nblocks; blk += nwaves) {
    const int row = blk * 16 + m;
    const int lab = label[row];
    const float* fr = feat    + (size_t)row * D_DIM + khalf * 2;
    const float* cr = centers + (size_t)lab * D_DIM + khalf * 2;
    #pragma unroll 8
    for (int kb = 0; kb < D_DIM; kb += 4) {
      v2f fa = *(const v2f*)(fr + kb);            // A chunk (b64 load)
      v2f cb = *(const v2f*)(cr + kb);            // B chunk (gathered, b64 load)
      // (neg_a, A, neg_b, B, c_mod, C, reuse_a, reuse_b)
      acc = __builtin_amdgcn_wmma_f32_16x16x4_f32(
          false, fa, false, cb, (short)0, acc, false, false);
      // each tile element visited exactly once across the 32 lanes:
      sf    += fa.x * fa.x + fa.y * fa.y;         // Sf partial
      sql   += cb.x * cb.x + cb.y * cb.y;         // SqL partial
      const int db = kb + khalf * 2;
      cross += fa.x * s_sumC[db] + fa.y * s_sumC[db + 1]; // crossAll partial
    }
  }

  // diagonal of D: VGPR v holds M=v at lane v and M=v+8 at lane v+24
  float g = 0.f;
  #pragma unroll
  for (int v = 0; v < 8; ++v) {
    float dv = acc[v];
    g += ((lane == v) || (lane == v + 24)) ? dv : 0.f;
  }

  for (int off = 16; off; off >>= 1) {
    sf    += __shfl_xor(sf,    off, 32);
    sql   += __shfl_xor(sql,   off, 32);
    cross += __shfl_xor(cross, off, 32);
    g     += __shfl_xor(g,     off, 32);
  }
  __shared__ float red[8][4];
  if (lane == 0) { red[wv][0] = sf; red[wv][1] = sql; red[wv][2] = cross; red[wv][3] = g; }
  __syncthreads();
  if (threadIdx.x < 4) {
    float s = 0.f;
    for (int w = 0; w < 8; ++w) s += red[w][threadIdx.x];
    ws[WS_FEATP + blockIdx.x * 4 + threadIdx.x] = s;
  }
}

// ---- Pass 4: scalar finish -------------------------------------------------
__global__ void ccl_final(const float* __restrict__ ws, float* __restrict__ out) {
  if (threadIdx.x == 0) {
    float Sf = 0.f, SqL = 0.f, crossAll = 0.f, G = 0.f;
    for (int w = 0; w < 128; ++w) {
      Sf       += ws[WS_FEATP + w * 4 + 0];
      SqL      += ws[WS_FEATP + w * 4 + 1];
      crossAll += ws[WS_FEATP + w * 4 + 2];
      G        += ws[WS_FEATP + w * 4 + 3];
    }
    const float Sc    = ws[WS_SC];
    const float total = (float)C_ROWS * Sf + (float)B_ROWS * Sc - 2.f * crossAll;
    const float intra = Sf + SqL - 2.f * G;
    const float inter = total - intra;
    out[0] = (1.0f / 2.0f / (float)B_ROWS) * intra / (inter + 1e-6f) / 0.1f;
  }
}

extern "C" void kernel_launch(void* const* d_in, const int* in_sizes, int n_in,
                              void* d_out, int out_size, void* d_ws, size_t ws_size,
                              hipStream_t stream) {
  const float* feat    = (const float*)d_in[0];
  const int*   label   = (const int*)d_in[1];   // harness delivers integer inputs as int32
  const float* centers = (const float*)d_in[2];
  float*       ws      = (float*)d_ws;          // ~19 KB used
  float*       out     = (float*)d_out;

  ccl_centers_partial<<<32, 256, 0, stream>>>(centers, ws);
  ccl_centers_reduce <<<1, 128, 0, stream>>>(ws);
  ccl_feat_pass      <<<128, 256, 0, stream>>>(feat, label, centers, ws);
  ccl_final          <<<1, 32, 0, stream>>>(ws, out);
}